// HungarianMatcher_87144886436644
// MI455X (gfx1250) — compile-verified
//
#include <hip/hip_runtime.h>
#include <hip/hip_bf16.h>
#include <stddef.h>

#define NB   64
#define NQ   256
#define NT   256
#define NCLS 256
#define OSTR (2 + NCLS)      // outputs last-dim stride: 2 pos + 256 logits
#define MAXD 100.0f
#define BIGF 1000000000.0f

typedef __attribute__((ext_vector_type(2))) float v2f;
typedef __attribute__((ext_vector_type(8))) float v8f;

// ---------------------------------------------------------------------------
// Kernel 1: per-(b,q) softmax stats (max, sum of exp) -> stats[bq*2 + {0,1}]
// ---------------------------------------------------------------------------
__global__ __launch_bounds__(256) void softmax_stats_kernel(
    const float* __restrict__ outputs, float* __restrict__ stats) {
  const int bq = blockIdx.x;                 // 0 .. NB*NQ-1
  const int t  = threadIdx.x;                // 0 .. 255 (one class per thread)
  const float* logits = outputs + (size_t)bq * OSTR + 2;
  __shared__ float red[256];

  const float x = logits[t];
  red[t] = x; __syncthreads();
  for (int s = 128; s > 0; s >>= 1) {
    if (t < s) red[t] = fmaxf(red[t], red[t + s]);
    __syncthreads();
  }
  const float m = red[0];
  __syncthreads();
  red[t] = __expf(x - m); __syncthreads();
  for (int s = 128; s > 0; s >>= 1) {
    if (t < s) red[t] += red[t + s];
    __syncthreads();
  }
  if (t == 0) { stats[2 * bq] = m; stats[2 * bq + 1] = red[0]; }
}

// ---------------------------------------------------------------------------
// Kernel 2: cost matrix C[b][t][q] via WMMA f32 16x16x4 (K=2 real, padded).
//   dot(t,q) = tgt_pos[t] . out_pos[q]   (A = tgt tile 16x4, B = outT 4x16)
//   dist^2   = |tgt|^2 + |out|^2 - 2*dot
//   C        = min(dist,100) (or 100 if id<=0)  - softmax(logits_q)[id_t]
// One wave per 16x16 (t,q) tile; 8 waves per block.
// ---------------------------------------------------------------------------
__global__ __launch_bounds__(256) void build_cost_kernel(
    const float* __restrict__ outputs, const float* __restrict__ tgt_pos,
    const int* __restrict__ tgt_ids, const float* __restrict__ stats,
    float* __restrict__ C) {
  const int b    = blockIdx.x;
  const int t0   = blockIdx.y * 16;
  const int wave = threadIdx.x >> 5;
  const int lane = threadIdx.x & 31;
  const int q0   = (blockIdx.z * 8 + wave) * 16;
  const int half = lane >> 4;      // 0: K=0,1   1: K=2,3 (zero padding)
  const int l16  = lane & 15;

  // A-matrix: 16x4 f32, lanes 0-15 hold K=0(v0),K=1(v1); lanes 16-31 K=2,3.
  v2f a; a.x = 0.0f; a.y = 0.0f;
  if (half == 0) {
    const float* tp = tgt_pos + 2 * ((size_t)b * NT + t0 + l16);
    a.x = tp[0]; a.y = tp[1];
  }
  // B-matrix: 4x16 f32, v0 holds rows K=0 (lanes 0-15) / K=2 (16-31), v1 K=1/K=3.
  v2f bm; bm.x = 0.0f; bm.y = 0.0f;
  {
    const float* op = outputs + ((size_t)b * NQ + q0 + l16) * OSTR;
    if (half == 0) { bm.x = op[0]; bm.y = op[1]; }
  }

  v8f acc = {};
  acc = __builtin_amdgcn_wmma_f32_16x16x4_f32(
      /*neg_a=*/false, a, /*neg_b=*/false, bm,
      /*c_mod=*/(short)0, acc, /*reuse_a=*/false, /*reuse_b=*/false);

  // C/D layout: VGPR r -> M = r + 8*half, N = lane&15.
  const int q = q0 + l16;
  const size_t qbase = ((size_t)b * NQ + q) * OSTR;
  const float qx = outputs[qbase + 0];
  const float qy = outputs[qbase + 1];
  const float qn = qx * qx + qy * qy;
  const float sm = stats[2 * ((size_t)b * NQ + q)];
  const float ss = stats[2 * ((size_t)b * NQ + q) + 1];
  const float* qlog = outputs + qbase + 2;

#pragma unroll
  for (int r = 0; r < 8; ++r) {
    const int t = t0 + r + half * 8;
    const float tx = tgt_pos[2 * ((size_t)b * NT + t) + 0];
    const float ty = tgt_pos[2 * ((size_t)b * NT + t) + 1];
    const int id = tgt_ids[(size_t)b * NT + t];
    const float d2 = tx * tx + ty * ty + qn - 2.0f * acc[r];
    const float dist = sqrtf(fmaxf(d2, 0.0f));
    float cb = fminf(dist, MAXD);
    if (id <= 0) cb = MAXD;
    const float cc = -__expf(qlog[id] - sm) / ss;
    C[((size_t)b * NT + t) * NQ + q] = cb + cc;
  }
}

// ---------------------------------------------------------------------------
// Kernel 3: Jonker-Volgenant shortest augmenting path, one block per batch.
// Columns j=0..255 map to threads. Per-column state (minv, used, dual v) is
// privatized into registers; argmin = 5 x shfl_xor (wave32) + 8-way LDS
// combine. 3 block barriers per Dijkstra step (critical path optimization).
// ---------------------------------------------------------------------------
__global__ __launch_bounds__(256) void hungarian_kernel(
    const float* __restrict__ Call, float* __restrict__ idx_out) {
  const int b = blockIdx.x;
  const float* cost = Call + (size_t)b * NT * NQ;
  const int j    = threadIdx.x;
  const int lane = j & 31;
  const int wv   = j >> 5;

  __shared__ float u[NT + 1];       // row duals (cross-thread)
  __shared__ int   p[NT + 1];       // column -> row matching (cross-thread)
  __shared__ int   way[NT];         // augmenting-path parents (read by t0)
  __shared__ float pv[8];
  __shared__ int   pi[8];
  __shared__ int   s_j0;
  __shared__ float s_delta;

  u[j] = 0.0f; p[j] = -1;
  if (j == 0) { u[NT] = 0.0f; p[NT] = -1; }
  float vj = 0.0f;                  // column dual v[j] (owner-only -> register)
  float vN = 0.0f;                  // v[NT], owned by thread 0
  __syncthreads();

  for (int i = 0; i < NT; ++i) {
    if (j == 0) { p[NT] = i; s_j0 = NT; }
    way[j] = NT;
    float mj   = BIGF;              // minv[j] (owner-only -> register)
    bool  usedj = false;            // used[j] (owner-only -> register)
    __builtin_prefetch(&cost[(size_t)i * NQ + j], 0, 3);  // global_prefetch_b8

    for (;;) {
      __syncthreads();              // (A) publish s_j0, u, p, way
      const int j0 = s_j0;
      if (p[j0] < 0) break;         // uniform exit (LDS-shared)
      const int   i0  = p[j0];
      const float ui0 = u[i0];
      if (j == j0) usedj = true;    // own flag; j0==NT handled by thread 0 below

      float key;
      if (!usedj) {
        const float cur = cost[(size_t)i0 * NQ + j] - ui0 - vj;
        if (cur < mj) { mj = cur; way[j] = j0; }
        key = mj;
      } else {
        key = BIGF;
      }
      int ki = j;
      // wave32 lexicographic argmin (first-index tie-break), no barriers
#pragma unroll
      for (int m = 16; m > 0; m >>= 1) {
        const float ov = __shfl_xor(key, m, 32);
        const int   oi = __shfl_xor(ki,  m, 32);
        if (ov < key || (ov == key && oi < ki)) { key = ov; ki = oi; }
      }
      if (lane == 0) { pv[wv] = key; pi[wv] = ki; }
      __syncthreads();              // (C) publish wave partials

      if (j == 0) {
        float d = pv[0]; int j1 = pi[0];
#pragma unroll
        for (int w = 1; w < 8; ++w) {
          if (pv[w] < d || (pv[w] == d && pi[w] < j1)) { d = pv[w]; j1 = pi[w]; }
        }
        s_delta = d; s_j0 = j1;
      }
      __syncthreads();              // (D) publish delta / next j0

      const float delta = s_delta;
      if (usedj) { u[p[j]] += delta; vj -= delta; }  // p injective on used cols
      else       { mj -= delta; }
      if (j == 0) { u[p[NT]] += delta; vN -= delta; } // column NT used from it.1
    }

    if (j == 0) {                   // augment along `way` (serial, short)
      int j0 = s_j0;
      while (j0 != NT) { const int jn = way[j0]; p[j0] = p[jn]; j0 = jn; }
    }
    __syncthreads();                // protect way[]/p[] before next row init
  }

  // indices[0][b][t] = t ; indices[1][b][t] = col assigned to row t
  idx_out[(size_t)b * NT + j] = (float)j;
  const int r = p[j];               // row matched to column j
  idx_out[(size_t)NB * NT + (size_t)b * NT + r] = (float)j;
}

// ---------------------------------------------------------------------------
// Launch: stats -> cost (WMMA) -> Hungarian. d_out = [indices(2*B*T) | C(B*T*Q)]
// ---------------------------------------------------------------------------
extern "C" void kernel_launch(void* const* d_in, const int* in_sizes, int n_in,
                              void* d_out, int out_size, void* d_ws, size_t ws_size,
                              hipStream_t stream) {
  (void)in_sizes; (void)n_in; (void)out_size; (void)ws_size;
  const float* outputs = (const float*)d_in[0];   // [B,Q,258]
  const float* tgt_pos = (const float*)d_in[1];   // [B,T,2]
  const int*   tgt_ids = (const int*)d_in[2];     // [B,T]

  float* out     = (float*)d_out;
  float* idx_out = out;                           // 2*B*T
  float* C       = out + 2 * NB * NT;             // B*T*Q
  float* stats   = (float*)d_ws;                  // B*Q*2 floats

  softmax_stats_kernel<<<NB * NQ, 256, 0, stream>>>(outputs, stats);
  build_cost_kernel<<<dim3(NB, NT / 16, 2), 256, 0, stream>>>(
      outputs, tgt_pos, tgt_ids, stats, C);
  hungarian_kernel<<<NB, 256, 0, stream>>>(C, idx_out);
}